// SparsePooling_2336462209101
// MI455X (gfx1250) — compile-verified
//
#include <hip/hip_runtime.h>
#include <hip/hip_bf16.h>

// ---- problem constants (match reference) ----
#define NVOX   2000000
#define CCH    32
#define GRIDX  128
#define BATCH  8
#define OUTG   64                               // (128 + 0 + 1) / 2
#define KEYSPACE (BATCH * OUTG * OUTG * OUTG)   // 8*64^3 = 2,097,152 = 2^21

// ---- scan config ----
#define SCAN_BLOCK 256
#define SCAN_ITEMS 8
#define SCAN_CHUNK (SCAN_BLOCK * SCAN_ITEMS)    // 2048 keys per block
#define SCAN_NBLK  (KEYSPACE / SCAN_CHUNK)      // 1024 blocks

__device__ __forceinline__ unsigned voxel_key(const int* __restrict__ coords, int i) {
    const int b = coords[4 * i + 0];
    const int x = coords[4 * i + 1] >> 1;   // PAD=0, STRIDE=2
    const int y = coords[4 * i + 2] >> 1;
    const int z = coords[4 * i + 3] >> 1;
    return (((unsigned)b * OUTG + (unsigned)x) * OUTG + (unsigned)y) * OUTG + (unsigned)z;
}

// K0: zero accumulators (vectorized b128 stores), zero count histogram,
// fill uniq-keys region with -1.
__global__ void sp_init_kernel(float* __restrict__ out, unsigned* __restrict__ cnt) {
    const long long stride = (long long)gridDim.x * blockDim.x;
    const long long gid = (long long)blockIdx.x * blockDim.x + threadIdx.x;
    float4* out4 = (float4*)out;                       // N*C = 64M floats, /4 = 16M float4
    const long long n4 = ((long long)NVOX * CCH) / 4;
    const float4 z4 = make_float4(0.f, 0.f, 0.f, 0.f);
    for (long long i = gid; i < n4; i += stride) out4[i] = z4;
    for (long long i = gid; i < KEYSPACE; i += stride) cnt[i] = 0u;
    float* uniq = out + (long long)NVOX * CCH;
    for (long long i = gid; i < NVOX; i += stride) uniq[i] = -1.0f;
}

// K1: histogram voxel counts per output cell (cnt is 8MB -> L2 resident)
__global__ void sp_count_kernel(const int* __restrict__ coords, unsigned* __restrict__ cnt) {
    const int i = blockIdx.x * blockDim.x + threadIdx.x;
    if (i >= NVOX) return;
    if (i + 1024 < NVOX) __builtin_prefetch(&coords[4 * (i + 1024)], 0, 0); // global_prefetch_b8
    const unsigned key = voxel_key(coords, i);
    atomicAdd(&cnt[key], 1u);
}

// K2a: per-block occupancy totals (occ = cnt>0), 2048 keys / block
__global__ void sp_scan1_kernel(const unsigned* __restrict__ cnt, unsigned* __restrict__ blk) {
    __shared__ unsigned sh[SCAN_BLOCK];
    const int tid = threadIdx.x;
    const int base = blockIdx.x * SCAN_CHUNK + tid * SCAN_ITEMS;
    unsigned local = 0;
    #pragma unroll
    for (int j = 0; j < SCAN_ITEMS; ++j) local += (cnt[base + j] > 0u) ? 1u : 0u;
    sh[tid] = local;
    __syncthreads();
    for (int off = SCAN_BLOCK >> 1; off > 0; off >>= 1) {
        if (tid < off) sh[tid] += sh[tid + off];
        __syncthreads();
    }
    if (tid == 0) blk[blockIdx.x] = sh[0];
}

// K2b: single-block exclusive scan of the 1024 block totals
__global__ void sp_scan2_kernel(unsigned* __restrict__ blk) {
    __shared__ unsigned sh[SCAN_NBLK];
    const int tid = threadIdx.x;   // blockDim.x == SCAN_NBLK == 1024
    sh[tid] = blk[tid];
    __syncthreads();
    for (int off = 1; off < SCAN_NBLK; off <<= 1) {
        unsigned v = (tid >= off) ? sh[tid - off] : 0u;
        __syncthreads();
        sh[tid] += v;
        __syncthreads();
    }
    blk[tid] = (tid == 0) ? 0u : sh[tid - 1];
}

// K2c: final scan pass: pref[key] = rank of key among occupied cells (ascending
// key order == jnp.unique sorted order); emit uniq[rank] = (float)key.
__global__ void sp_scan3_kernel(const unsigned* __restrict__ cnt,
                                const unsigned* __restrict__ blk,
                                unsigned* __restrict__ pref,
                                float* __restrict__ uniq) {
    __shared__ unsigned sh[SCAN_BLOCK];
    const int tid = threadIdx.x;
    const int base = blockIdx.x * SCAN_CHUNK + tid * SCAN_ITEMS;
    unsigned occ[SCAN_ITEMS];
    unsigned local = 0;
    #pragma unroll
    for (int j = 0; j < SCAN_ITEMS; ++j) {
        occ[j] = (cnt[base + j] > 0u) ? 1u : 0u;
        local += occ[j];
    }
    sh[tid] = local;
    __syncthreads();
    for (int off = 1; off < SCAN_BLOCK; off <<= 1) {
        unsigned v = (tid >= off) ? sh[tid - off] : 0u;
        __syncthreads();
        sh[tid] += v;
        __syncthreads();
    }
    const unsigned texcl = (tid == 0) ? 0u : sh[tid - 1];
    unsigned run = blk[blockIdx.x] + texcl;
    #pragma unroll
    for (int j = 0; j < SCAN_ITEMS; ++j) {
        const unsigned k = (unsigned)(base + j);
        pref[k] = run;
        if (occ[j]) {
            uniq[run] = (float)k;   // key < 2^21 -> exact in f32
            ++run;
        }
    }
}

// K3: one wave32 per voxel, one lane per channel. Lane 0 resolves the
// key->rank lookup once per wave; broadcast via v_readfirstlane (wave32).
// Then a coalesced 128B burst of global_atomic_add_f32 into the pooled row.
__global__ void sp_scatter_kernel(const float* __restrict__ feats,
                                  const int* __restrict__ coords,
                                  const unsigned* __restrict__ pref,
                                  float* __restrict__ out) {
    const long long gid = (long long)blockIdx.x * blockDim.x + threadIdx.x;
    const int i = (int)(gid >> 5);
    const int lane = (int)(gid & 31);
    if (i >= NVOX) return;
    unsigned r = 0;
    if (lane == 0) {
        if (i + 64 < NVOX) __builtin_prefetch(&coords[4 * (i + 64)], 0, 0);
        const unsigned key = voxel_key(coords, i);
        r = pref[key];
    }
    r = (unsigned)__builtin_amdgcn_readfirstlane((int)r);   // wave-uniform rank
    atomicAdd(&out[(long long)r * CCH + lane], feats[(long long)i * CCH + lane]);
}

// K4: average -> divide each occupied pooled row by its voxel count.
// Lane 0 resolves uniq/cnt once per wave; rows >= M (uniq == -1) stay zero,
// matching the reference's padded rows.
__global__ void sp_finalize_kernel(const unsigned* __restrict__ cnt,
                                   const float* __restrict__ uniq,
                                   float* __restrict__ out) {
    const long long gid = (long long)blockIdx.x * blockDim.x + threadIdx.x;
    const int r = (int)(gid >> 5);
    const int lane = (int)(gid & 31);
    if (r >= NVOX) return;
    int c = 0;
    if (lane == 0) {
        const float kf = uniq[r];
        if (kf >= 0.0f) c = (int)cnt[(unsigned)kf];
    }
    c = __builtin_amdgcn_readfirstlane(c);                  // wave-uniform count
    if (c == 0) return;                                     // padded / empty row
    out[(long long)r * CCH + lane] /= (float)c;
}

extern "C" void kernel_launch(void* const* d_in, const int* in_sizes, int n_in,
                              void* d_out, int out_size, void* d_ws, size_t ws_size,
                              hipStream_t stream) {
    const float* feats  = (const float*)d_in[0];   // [N,32] f32
    const int*   coords = (const int*)d_in[1];     // [N,4]  i32
    float* out  = (float*)d_out;                   // [N*32] feats then [N] keys-as-f32
    float* uniq = out + (long long)NVOX * CCH;

    // workspace: cnt[2^21] u32 | pref[2^21] u32 | blk[1024] u32  (~16.8 MB)
    unsigned* cnt  = (unsigned*)d_ws;
    unsigned* pref = cnt + KEYSPACE;
    unsigned* blk  = pref + KEYSPACE;
    (void)in_sizes; (void)n_in; (void)out_size; (void)ws_size;

    sp_init_kernel<<<4096, 256, 0, stream>>>(out, cnt);

    sp_count_kernel<<<(NVOX + 255) / 256, 256, 0, stream>>>(coords, cnt);

    sp_scan1_kernel<<<SCAN_NBLK, SCAN_BLOCK, 0, stream>>>(cnt, blk);
    sp_scan2_kernel<<<1, SCAN_NBLK, 0, stream>>>(blk);
    sp_scan3_kernel<<<SCAN_NBLK, SCAN_BLOCK, 0, stream>>>(cnt, blk, pref, uniq);

    const long long scat_threads = (long long)NVOX * 32;
    sp_scatter_kernel<<<(unsigned)((scat_threads + 255) / 256), 256, 0, stream>>>(
        feats, coords, pref, out);

    sp_finalize_kernel<<<(unsigned)((scat_threads + 255) / 256), 256, 0, stream>>>(
        cnt, uniq, out);
}